// social_stgcnn24_17875653886713
// MI455X (gfx1250) — compile-verified
//
#include <hip/hip_runtime.h>
#include <math.h>

#define V_N 2000
#define BN_EPS 1e-5f
#define KC 10           // K-split chunks for the big einsum
#define KCHUNK 200      // 2000 / KC, multiple of 8
#define NIT (KCHUNK / 8)

typedef __attribute__((ext_vector_type(2))) float v2f;
typedef __attribute__((ext_vector_type(8))) float v8f;

// ---------------------------------------------------------------------------
// Kernel 0: 1x1 convs. Writes xg zero-PADDED to 16 rows (WMMA M-pad) so the
// hot loop needs no masking, plus residual branch incl. BN.
// ---------------------------------------------------------------------------
__global__ void k_pre(const float* __restrict__ v,
                      const float* __restrict__ gcn_w, const float* __restrict__ gcn_b,
                      const float* __restrict__ res_w, const float* __restrict__ res_b,
                      const float* __restrict__ bnr_g, const float* __restrict__ bnr_b,
                      const float* __restrict__ bnr_m, const float* __restrict__ bnr_v,
                      float* __restrict__ xgp, float* __restrict__ resb) {
  int idx = blockIdx.x * blockDim.x + threadIdx.x;
  if (idx >= 8 * V_N) return;
  int t = idx / V_N, vv = idx % V_N;
  float v0 = v[(0 * 8 + t) * V_N + vv];
  float v1 = v[(1 * 8 + t) * V_N + vv];
#pragma unroll
  for (int c = 0; c < 5; ++c) {
    xgp[(c * 8 + t) * V_N + vv] = gcn_w[c * 2 + 0] * v0 + gcn_w[c * 2 + 1] * v1 + gcn_b[c];
    float rc = res_w[c * 2 + 0] * v0 + res_w[c * 2 + 1] * v1 + res_b[c];
    float sc = rsqrtf(bnr_v[c] + BN_EPS) * bnr_g[c];
    resb[(c * 8 + t) * V_N + vv] = (rc - bnr_m[c]) * sc + bnr_b[c];
  }
#pragma unroll
  for (int m = 5; m < 16; ++m) xgp[(m * 8 + t) * V_N + vv] = 0.f;
}

// ---------------------------------------------------------------------------
// Kernel 1: heavy einsum y[c,t,w] = sum_v xg[c,t,v]*a[t,v,w] via WMMA f32,
// fused with the 128MB copy of `a` into the output tuple region.
// K split into KC chunks: 8*125*KC = 10000 waves. Scalar (readfirstlane)
// chunk bounds -> uniform loop; loads use scalar base + invariant lane offset
// (SADDR form, constant immediates). Depth-2 SW pipeline.
// ---------------------------------------------------------------------------
__device__ __forceinline__ void ld_stage(const float* __restrict__ A,  // a + t*V*V (scalar)
                                         const float* __restrict__ xg, // xgp (scalar)
                                         int k,                        // scalar stage base
                                         int off0, int offA,          // per-lane, loop-invariant
                                         float* b, v2f* av) {
  const float* Bk = A + (size_t)k * V_N;
  const float* Xk = xg + k;
  b[0] = Bk[off0];
  b[1] = Bk[off0 + V_N];
  b[2] = Bk[off0 + 4 * V_N];
  b[3] = Bk[off0 + 5 * V_N];
  av[0] = *(const v2f*)&Xk[offA];       // 8B aligned (offA, k even)
  av[1] = *(const v2f*)&Xk[offA + 4];
}

__global__ void __launch_bounds__(256) k_gcn_wmma(const float* __restrict__ xgp,
                                                  const float* __restrict__ A,
                                                  float* __restrict__ ypart,
                                                  float* __restrict__ a_copy) {
  const int V = V_N;
  // wave-uniform, forced scalar so loop bounds/addresses live in SGPRs
  int waveId = __builtin_amdgcn_readfirstlane(
      blockIdx.x * (blockDim.x >> 5) + (threadIdx.x >> 5));  // 0..9999
  int lane = threadIdx.x & 31;
  int kc   = waveId % KC;
  int tile = waveId / KC;          // 0..999
  int t    = tile / 125;
  int n0   = (tile % 125) * 16;
  int half = lane >> 4;            // K-pair selector (A/B), M-half (C/D)
  int nl   = lane & 15;            // N column (B/C/D), M row (A)
  int col  = n0 + nl;

  const float* __restrict__ At = A + (size_t)t * V * V;       // scalar base
  float* __restrict__       Ct = a_copy + (size_t)t * V * V;  // scalar base
  const int off0 = 2 * half * V + col;           // per-lane, invariant (B / copy)
  const int offA = (nl * 8 + t) * V + 2 * half;  // per-lane, invariant (A, padded rows)

  const int kbeg = kc * KCHUNK;
  const int kend = kbeg + KCHUNK;

  v8f acc = {0.f, 0.f, 0.f, 0.f, 0.f, 0.f, 0.f, 0.f};

  // depth-2 pipeline: slots s0 (consume now) and s1 (consume next)
  float b0[4], b1[4];
  v2f   a0[2], a1[2];
  ld_stage(At, xgp, kbeg, off0, offA, b0, a0);
  ld_stage(At, xgp, kbeg + 8, off0, offA, b1, a1);

#pragma unroll 2
  for (int it = 0; it < NIT; ++it) {
    int k = kbeg + it * 8;                              // scalar
    // issue stage it+2 loads (dummy reload of kbeg on the tail; scalar select)
    int k2 = (it + 2 < NIT) ? (k + 16) : kbeg;
    float nb[4];
    v2f   na[2];
    ld_stage(At, xgp, k2, off0, offA, nb, na);
    // branchless scalar-clamped prefetch of the B stream ~16 stages ahead
    int kp = (k + 128 < kend) ? (k + 128) : kbeg;
    __builtin_prefetch(At + (size_t)kp * V + off0, 0, 1);

    // fused copy of `a` (each element touched exactly once across all waves)
    float* Ck = Ct + (size_t)k * V;
    Ck[off0]         = b0[0];
    Ck[off0 + V]     = b0[1];
    Ck[off0 + 4 * V] = b0[2];
    Ck[off0 + 5 * V] = b0[3];

    v2f bv0, bv1;
    bv0.x = b0[0]; bv0.y = b0[1];
    bv1.x = b0[2]; bv1.y = b0[3];
    acc = __builtin_amdgcn_wmma_f32_16x16x4_f32(false, a0[0], false, bv0,
                                                (short)0, acc, false, false);
    acc = __builtin_amdgcn_wmma_f32_16x16x4_f32(false, a0[1], false, bv1,
                                                (short)0, acc, false, false);
    // rotate slots (renamed away by unroll-2)
    b0[0] = b1[0]; b0[1] = b1[1]; b0[2] = b1[2]; b0[3] = b1[3];
    a0[0] = a1[0]; a0[1] = a1[1];
    b1[0] = nb[0]; b1[1] = nb[1]; b1[2] = nb[2]; b1[3] = nb[3];
    a1[0] = na[0]; a1[1] = na[1];
  }

  // C/D layout: lanes 0-15 hold M=0..7 in VGPR0..7; rows 0..4 are real channels
  if (half == 0) {
    float* yp = ypart + (size_t)kc * 40 * V;
#pragma unroll
    for (int c = 0; c < 5; ++c) yp[(c * 8 + t) * V + col] = acc[c];
  }
}

// ---------------------------------------------------------------------------
// Kernel 1b: reduce the KC partial sums
// ---------------------------------------------------------------------------
__global__ void k_red(const float* __restrict__ ypart, float* __restrict__ y) {
  int i = blockIdx.x * blockDim.x + threadIdx.x;
  if (i >= 40 * V_N) return;
  float s = 0.f;
#pragma unroll
  for (int kc = 0; kc < KC; ++kc) s += ypart[(size_t)kc * 40 * V_N + i];
  y[i] = s;
}

// ---------------------------------------------------------------------------
// Kernel 2: BN1 + PReLU + temporal conv (3,1) pad(1,0) + BN2 + residual + PReLU
// ---------------------------------------------------------------------------
__global__ void k_stgcn_post(const float* __restrict__ y, const float* __restrict__ resb,
                             const float* __restrict__ bn1_g, const float* __restrict__ bn1_b,
                             const float* __restrict__ bn1_m, const float* __restrict__ bn1_v,
                             const float* __restrict__ prelu_tcn,
                             const float* __restrict__ tcn_w, const float* __restrict__ tcn_b,
                             const float* __restrict__ bn2_g, const float* __restrict__ bn2_b,
                             const float* __restrict__ bn2_m, const float* __restrict__ bn2_v,
                             const float* __restrict__ prelu_st,
                             float* __restrict__ st) {
  int idx = blockIdx.x * blockDim.x + threadIdx.x;
  if (idx >= 8 * V_N) return;
  int t = idx / V_N, vv = idx % V_N;
  float a1 = prelu_tcn[0], a2 = prelu_st[0];

  float x1[5][3];
#pragma unroll
  for (int cp = 0; cp < 5; ++cp) {
    float sc = rsqrtf(bn1_v[cp] + BN_EPS) * bn1_g[cp];
    float sh = bn1_b[cp] - bn1_m[cp] * sc;
#pragma unroll
    for (int dt = 0; dt < 3; ++dt) {
      int tt = t + dt - 1;
      float val = 0.f;
      if (tt >= 0 && tt < 8) {
        val = y[(cp * 8 + tt) * V_N + vv] * sc + sh;
        val = val >= 0.f ? val : a1 * val;  // PReLU applied before conv zero-pad
      }
      x1[cp][dt] = val;
    }
  }
#pragma unroll
  for (int c = 0; c < 5; ++c) {
    float s = tcn_b[c];
#pragma unroll
    for (int cp = 0; cp < 5; ++cp)
#pragma unroll
      for (int dt = 0; dt < 3; ++dt) s += x1[cp][dt] * tcn_w[(c * 5 + cp) * 3 + dt];
    float sc = rsqrtf(bn2_v[c] + BN_EPS) * bn2_g[c];
    s = (s - bn2_m[c]) * sc + bn2_b[c];
    s += resb[(c * 8 + t) * V_N + vv];
    st[(c * 8 + t) * V_N + vv] = s >= 0.f ? s : a2 * s;
  }
}

// ---------------------------------------------------------------------------
// Kernel 3: tp0 3x3 conv (8->12 over (5,V), pad 1) + PReLU; weights in LDS.
// Input = st buffer reinterpreted (raw torch .view) as (8,5,V).
// ---------------------------------------------------------------------------
__global__ void __launch_bounds__(256) k_tp0(const float* __restrict__ in,
                                             const float* __restrict__ w,
                                             const float* __restrict__ b,
                                             const float* __restrict__ prelu0,
                                             float* __restrict__ v2a) {
  __shared__ float sw[864];
  __shared__ float sb[13];
  for (int i = threadIdx.x; i < 864; i += 256) sw[i] = w[i];
  if (threadIdx.x < 12) sb[threadIdx.x] = b[threadIdx.x];
  if (threadIdx.x == 12) sb[12] = prelu0[0];
  __syncthreads();

  int idx = blockIdx.x * blockDim.x + threadIdx.x;
  if (idx >= 5 * V_N) return;
  int h = idx / V_N, x = idx % V_N;
  float al = sb[12];
  float acc[12];
#pragma unroll
  for (int o = 0; o < 12; ++o) acc[o] = sb[o];
  for (int i = 0; i < 8; ++i) {
    float in9[9];
#pragma unroll
    for (int kh = 0; kh < 3; ++kh)
#pragma unroll
      for (int kw = 0; kw < 3; ++kw) {
        int hh = h + kh - 1, ww = x + kw - 1;
        in9[kh * 3 + kw] = (hh >= 0 && hh < 5 && ww >= 0 && ww < V_N)
                               ? in[(i * 5 + hh) * V_N + ww] : 0.f;
      }
#pragma unroll
    for (int o = 0; o < 12; ++o)
#pragma unroll
      for (int kk = 0; kk < 9; ++kk) acc[o] += in9[kk] * sw[(o * 8 + i) * 9 + kk];
  }
#pragma unroll
  for (int o = 0; o < 12; ++o) {
    float s = acc[o];
    v2a[(o * 5 + h) * V_N + x] = s >= 0.f ? s : al * s;
  }
}

// ---------------------------------------------------------------------------
// Kernel 4a: dictionary projections kd = dict@W2^T+b2, hd = dict@W3^T+b3
// ---------------------------------------------------------------------------
__global__ void k_kh(const float* __restrict__ dict,
                     const float* __restrict__ W2, const float* __restrict__ b2,
                     const float* __restrict__ W3, const float* __restrict__ b3,
                     float* __restrict__ kd, float* __restrict__ hd) {
  int idx = blockIdx.x * blockDim.x + threadIdx.x;  // ((s*32)+d)*16+j
  if (idx >= 5 * 32 * 16) return;
  int j = idx & 15, sd = idx >> 4;
  const float* r = dict + sd * 12;
  float k = b2[j], h = b3[j];
#pragma unroll
  for (int c = 0; c < 12; ++c) {
    k += W2[j * 12 + c] * r[c];
    h += W3[j * 12 + c] * r[c];
  }
  kd[idx] = k;
  hd[idx] = h;
}

// ---------------------------------------------------------------------------
// Kernel 4: dictionary attention, one thread per (s, v); kd/hd staged in LDS
// ---------------------------------------------------------------------------
__global__ void __launch_bounds__(256) k_attn(const float* __restrict__ v2a,
                                              const float* __restrict__ W1,
                                              const float* __restrict__ b1,
                                              const float* __restrict__ kd,
                                              const float* __restrict__ hd,
                                              const float* __restrict__ W4,
                                              const float* __restrict__ b4,
                                              const float* __restrict__ ls_g,
                                              float* __restrict__ v2b) {
  __shared__ float skd[2560];
  __shared__ float shd[2560];
  for (int i = threadIdx.x; i < 2560; i += 256) {
    skd[i] = kd[i];
    shd[i] = hd[i];
  }
  __syncthreads();

  int idx = blockIdx.x * blockDim.x + threadIdx.x;
  if (idx >= 5 * V_N) return;
  int s = idx / V_N, vv = idx % V_N;
  float r[12];
#pragma unroll
  for (int c = 0; c < 12; ++c) r[c] = v2a[(c * 5 + s) * V_N + vv];
  float q[16];
#pragma unroll
  for (int j = 0; j < 16; ++j) {
    float t = b1[j];
#pragma unroll
    for (int c = 0; c < 12; ++c) t += W1[j * 12 + c] * r[c];
    q[j] = t;
  }
  float sc[32], mx = -1e30f;
  for (int d = 0; d < 32; ++d) {
    const float* kr = skd + (s * 32 + d) * 16;
    float t = 0.f;
#pragma unroll
    for (int j = 0; j < 16; ++j) t += q[j] * kr[j];
    t *= 0.25f;  // 1/sqrt(VAR1=16)
    sc[d] = t;
    mx = fmaxf(mx, t);
  }
  float sum = 0.f;
  for (int d = 0; d < 32; ++d) { sc[d] = expf(sc[d] - mx); sum += sc[d]; }
  float inv = 1.f / sum;
  float rr[16];
#pragma unroll
  for (int j = 0; j < 16; ++j) rr[j] = 0.f;
  for (int d = 0; d < 32; ++d) {
    float wgt = sc[d] * inv;
    const float* hr = shd + (s * 32 + d) * 16;
#pragma unroll
    for (int j = 0; j < 16; ++j) rr[j] += wgt * hr[j];
  }
#pragma unroll
  for (int c = 0; c < 12; ++c) {
    float o = b4[c];
#pragma unroll
    for (int j = 0; j < 16; ++j) o += W4[c * 16 + j] * rr[j];
    v2b[(c * 5 + s) * V_N + vv] = r[c] + ls_g[c] * o;
  }
}

// ---------------------------------------------------------------------------
// Kernel 5: gated unit, one thread per (s, v)
// ---------------------------------------------------------------------------
__global__ void k_gated(const float* __restrict__ v2b,
                        const float* __restrict__ Wu, const float* __restrict__ bu,
                        const float* __restrict__ Wv, const float* __restrict__ bv,
                        const float* __restrict__ Wo, const float* __restrict__ bo,
                        const float* __restrict__ ls2_g, float* __restrict__ v2c) {
  int idx = blockIdx.x * blockDim.x + threadIdx.x;
  if (idx >= 5 * V_N) return;
  int s = idx / V_N, vv = idx % V_N;
  float r[12];
#pragma unroll
  for (int c = 0; c < 12; ++c) r[c] = v2b[(c * 5 + s) * V_N + vv];
  float g[16];
#pragma unroll
  for (int j = 0; j < 16; ++j) {
    float u1 = bu[j], u2 = bv[j];
#pragma unroll
    for (int c = 0; c < 12; ++c) {
      u1 += Wu[j * 12 + c] * r[c];
      u2 += Wv[j * 12 + c] * r[c];
    }
    g[j] = u1 * (1.f / (1.f + expf(-u2)));
  }
#pragma unroll
  for (int c = 0; c < 12; ++c) {
    float o = bo[c];
#pragma unroll
    for (int j = 0; j < 16; ++j) o += Wo[c * 16 + j] * g[j];
    v2c[(c * 5 + s) * V_N + vv] = r[c] + ls2_g[c] * o;
  }
}

// ---------------------------------------------------------------------------
// Kernel 6: tpo 3x3 conv (12->12) -> out; weights in LDS. Final torch .view to
// (1,5,12,V) is a raw reinterpret of the same (12,5,V) buffer.
// ---------------------------------------------------------------------------
__global__ void __launch_bounds__(256) k_tpo(const float* __restrict__ in,
                                             const float* __restrict__ w,
                                             const float* __restrict__ b,
                                             float* __restrict__ out) {
  __shared__ float sw[1296];
  __shared__ float sb[12];
  for (int i = threadIdx.x; i < 1296; i += 256) sw[i] = w[i];
  if (threadIdx.x < 12) sb[threadIdx.x] = b[threadIdx.x];
  __syncthreads();

  int idx = blockIdx.x * blockDim.x + threadIdx.x;
  if (idx >= 5 * V_N) return;
  int h = idx / V_N, x = idx % V_N;
  float acc[12];
#pragma unroll
  for (int o = 0; o < 12; ++o) acc[o] = sb[o];
  for (int i = 0; i < 12; ++i) {
    float in9[9];
#pragma unroll
    for (int kh = 0; kh < 3; ++kh)
#pragma unroll
      for (int kw = 0; kw < 3; ++kw) {
        int hh = h + kh - 1, ww = x + kw - 1;
        in9[kh * 3 + kw] = (hh >= 0 && hh < 5 && ww >= 0 && ww < V_N)
                               ? in[(i * 5 + hh) * V_N + ww] : 0.f;
      }
#pragma unroll
    for (int o = 0; o < 12; ++o)
#pragma unroll
      for (int kk = 0; kk < 9; ++kk) acc[o] += in9[kk] * sw[(o * 12 + i) * 9 + kk];
  }
#pragma unroll
  for (int o = 0; o < 12; ++o) out[(o * 5 + h) * V_N + x] = acc[o];
}

// ---------------------------------------------------------------------------
extern "C" void kernel_launch(void* const* d_in, const int* in_sizes, int n_in,
                              void* d_out, int out_size, void* d_ws, size_t ws_size,
                              hipStream_t stream) {
  const float* v   = (const float*)d_in[0];
  const float* a   = (const float*)d_in[1];
  const float* gcn_w = (const float*)d_in[2];
  const float* gcn_b = (const float*)d_in[3];
  const float* bn1_g = (const float*)d_in[4];
  const float* bn1_b = (const float*)d_in[5];
  const float* bn1_m = (const float*)d_in[6];
  const float* bn1_v = (const float*)d_in[7];
  const float* prelu_tcn = (const float*)d_in[8];
  const float* tcn_w = (const float*)d_in[9];
  const float* tcn_b = (const float*)d_in[10];
  const float* bn2_g = (const float*)d_in[11];
  const float* bn2_b = (const float*)d_in[12];
  const float* bn2_m = (const float*)d_in[13];
  const float* bn2_v = (const float*)d_in[14];
  const float* res_w = (const float*)d_in[15];
  const float* res_b = (const float*)d_in[16];
  const float* bnr_g = (const float*)d_in[17];
  const float* bnr_b = (const float*)d_in[18];
  const float* bnr_m = (const float*)d_in[19];
  const float* bnr_v = (const float*)d_in[20];
  const float* prelu_st = (const float*)d_in[21];
  const float* tp0_w = (const float*)d_in[22];
  const float* tp0_b = (const float*)d_in[23];
  const float* prelu0 = (const float*)d_in[24];
  const float* tpo_w = (const float*)d_in[25];
  const float* tpo_b = (const float*)d_in[26];
  const float* hdict = (const float*)d_in[27];
  const float* W1 = (const float*)d_in[28];
  const float* b1 = (const float*)d_in[29];
  const float* W2 = (const float*)d_in[30];
  const float* b2 = (const float*)d_in[31];
  const float* W3 = (const float*)d_in[32];
  const float* b3 = (const float*)d_in[33];
  const float* W4 = (const float*)d_in[34];
  const float* b4 = (const float*)d_in[35];
  const float* ls_g  = (const float*)d_in[36];
  const float* ls2_g = (const float*)d_in[37];
  const float* Wu = (const float*)d_in[38];
  const float* bu = (const float*)d_in[39];
  const float* Wv = (const float*)d_in[40];
  const float* bv = (const float*)d_in[41];
  const float* Wo = (const float*)d_in[42];
  const float* bo = (const float*)d_in[43];

  float* out    = (float*)d_out;
  float* a_copy = out + 12 * 5 * V_N;   // tuple: (out[120000], a[32M])

  float* ws    = (float*)d_ws;
  float* xgp   = ws;                    // 16*8*V = 256000 (M-padded A rows)
  float* resb  = ws + 256000;           // 40*V  =  80000
  float* ypart = ws + 336000;           // KC*40*V = 800000
  float* y     = ws + 1136000;          // 40*V
  float* st    = ws + 1216000;          // 40*V (post st_gcn, viewed as (8,5,V))
  float* v2a   = ws + 1296000;          // 60*V
  float* v2b   = ws + 1416000;          // 60*V
  float* v2c   = ws + 1536000;          // 60*V
  float* kd    = ws + 1656000;          // 2560
  float* hd    = ws + 1658560;          // 2560

  const int B = 256;
  k_pre<<<(8 * V_N + B - 1) / B, B, 0, stream>>>(v, gcn_w, gcn_b, res_w, res_b,
                                                 bnr_g, bnr_b, bnr_m, bnr_v, xgp, resb);
  k_gcn_wmma<<<(8 * 125 * KC) / 8, B, 0, stream>>>(xgp, a, ypart, a_copy);  // 1250 blocks
  k_red<<<(40 * V_N + B - 1) / B, B, 0, stream>>>(ypart, y);
  k_stgcn_post<<<(8 * V_N + B - 1) / B, B, 0, stream>>>(y, resb,
      bn1_g, bn1_b, bn1_m, bn1_v, prelu_tcn, tcn_w, tcn_b,
      bn2_g, bn2_b, bn2_m, bn2_v, prelu_st, st);
  k_tp0<<<(5 * V_N + B - 1) / B, B, 0, stream>>>(st, tp0_w, tp0_b, prelu0, v2a);
  k_kh<<<(2560 + B - 1) / B, B, 0, stream>>>(hdict, W2, b2, W3, b3, kd, hd);
  k_attn<<<(5 * V_N + B - 1) / B, B, 0, stream>>>(v2a, W1, b1, kd, hd, W4, b4, ls_g, v2b);
  k_gated<<<(5 * V_N + B - 1) / B, B, 0, stream>>>(v2b, Wu, bu, Wv, bv, Wo, bo, ls2_g, v2c);
  k_tpo<<<(5 * V_N + B - 1) / B, B, 0, stream>>>(v2c, tpo_w, tpo_b, out);
}